// GNN_82583631167933
// MI455X (gfx1250) — compile-verified
//
#include <hip/hip_runtime.h>
#include <hip/hip_bf16.h>

typedef __attribute__((ext_vector_type(2))) float v2f;
typedef __attribute__((ext_vector_type(8))) float v8f;

#define FDIM 64

// ---------------------------------------------------------------- zero fill
__global__ void gnn_zero_kernel(float* __restrict__ p, int n) {
    int i = blockIdx.x * blockDim.x + threadIdx.x;
    int stride = gridDim.x * blockDim.x;
    for (; i < n; i += stride) p[i] = 0.0f;
}

// ------------------------------------------------------- edge scatter (sum)
// work item = (edge, 4-float chunk). 16 chunks per edge (64 feats).
__global__ void gnn_scatter_kernel(const float* __restrict__ feat,
                                   const long long* __restrict__ ei,
                                   float* __restrict__ agg, int n_edges) {
    int tid = blockIdx.x * blockDim.x + threadIdx.x;
    int total = n_edges * 16;
    if (tid >= total) return;
    int e = tid >> 4;
    int c = (tid & 15) * 4;
    long long s = ei[e];
    long long d = ei[n_edges + e];
    const float4 v = *(const float4*)(feat + (size_t)s * FDIM + c);
    float* dst = agg + (size_t)d * FDIM + c;
    unsafeAtomicAdd(dst + 0, v.x);
    unsafeAtomicAdd(dst + 1, v.y);
    unsafeAtomicAdd(dst + 2, v.z);
    unsafeAtomicAdd(dst + 3, v.w);
}

// -------------------------------------------- fused GraphConv linear (WMMA)
// out = act( agg @ Wrel^T + brel + xin @ Wroot^T )
// Wave handles one 16x16 output tile; block = 4 waves covering N=64.
// f32 WMMA 16x16x4: A 16x4 (2 VGPR), B 4x16 (2 VGPR), C/D 16x16 (8 VGPR).
__global__ __launch_bounds__(128)
void gnn_gemm_kernel(const float* __restrict__ agg, const float* __restrict__ xin,
                     const float* __restrict__ Wrel, const float* __restrict__ brel,
                     const float* __restrict__ Wroot, float* __restrict__ out,
                     int n_nodes, int do_relu) {
    const int wave = threadIdx.x >> 5;       // N-tile 0..3
    const int lane = threadIdx.x & 31;
    const int lm   = lane & 15;
    const int hi   = lane >> 4;              // 0: K=0,1 half; 1: K=2,3 half
    const int m0   = blockIdx.x * 16;
    const int n0   = wave * 16;
    const int ncol = n0 + lm;                // output feature this lane owns (B/C/D)

    // B fragments for both weight matrices, resident in registers.
    // B[k][n] = W[n*64 + k] (W is [out,in] row-major, we need W^T as KxN).
    v2f bw[16], bu[16];
#pragma unroll
    for (int kk = 0; kk < 16; ++kk) {
        int kb = kk * 4 + hi * 2;
        bw[kk].x = Wrel [ncol * FDIM + kb];
        bw[kk].y = Wrel [ncol * FDIM + kb + 1];
        bu[kk].x = Wroot[ncol * FDIM + kb];
        bu[kk].y = Wroot[ncol * FDIM + kb + 1];
    }

    int mrow = m0 + lm;
    if (mrow >= n_nodes) mrow = n_nodes - 1;   // clamp loads; EXEC stays all-1s
    const float* aggR = agg + (size_t)mrow * FDIM;
    const float* xinR = xin + (size_t)mrow * FDIM;

    v8f acc = {};
#pragma unroll
    for (int kk = 0; kk < 16; ++kk) {
        int kb = kk * 4 + hi * 2;
        v2f a; a.x = aggR[kb]; a.y = aggR[kb + 1];
        acc = __builtin_amdgcn_wmma_f32_16x16x4_f32(false, a, false, bw[kk],
                                                    (short)0, acc, false, false);
    }
#pragma unroll
    for (int kk = 0; kk < 16; ++kk) {
        int kb = kk * 4 + hi * 2;
        v2f a; a.x = xinR[kb]; a.y = xinR[kb + 1];
        acc = __builtin_amdgcn_wmma_f32_16x16x4_f32(false, a, false, bu[kk],
                                                    (short)0, acc, false, false);
    }

    const float bias = brel[ncol];
#pragma unroll
    for (int r = 0; r < 8; ++r) {
        int row = m0 + r + hi * 8;           // C/D layout: VGPR r -> rows r, r+8
        float v = acc[r] + bias;
        if (do_relu) v = fmaxf(v, 0.0f);
        if (row < n_nodes) out[(size_t)row * FDIM + ncol] = v;
    }
}

// --------------------------------------------------- segment sum per graph
__global__ void gnn_pool_kernel(const float* __restrict__ h,
                                const long long* __restrict__ batch,
                                float* __restrict__ psum, float* __restrict__ cnt,
                                int n_nodes) {
    int tid = blockIdx.x * blockDim.x + threadIdx.x;
    int total = n_nodes * 16;
    if (tid >= total) return;
    int n = tid >> 4;
    int c = (tid & 15) * 4;
    long long g = batch[n];
    const float4 v = *(const float4*)(h + (size_t)n * FDIM + c);
    float* d = psum + (size_t)g * FDIM + c;
    unsafeAtomicAdd(d + 0, v.x);
    unsafeAtomicAdd(d + 1, v.y);
    unsafeAtomicAdd(d + 2, v.z);
    unsafeAtomicAdd(d + 3, v.w);
    if ((tid & 15) == 0) unsafeAtomicAdd(cnt + g, 1.0f);
}

// ------------------------------------------------------------- final linear
__global__ void gnn_final_kernel(const float* __restrict__ psum,
                                 const float* __restrict__ cnt,
                                 const float* __restrict__ Wlin,
                                 const float* __restrict__ blin,
                                 float* __restrict__ out, int n_graphs) {
    int tid = blockIdx.x * blockDim.x + threadIdx.x;
    if (tid >= n_graphs * 2) return;
    int g = tid >> 1, o = tid & 1;
    float inv = 1.0f / fmaxf(cnt[g], 1.0f);
    float s = 0.0f;
#pragma unroll
    for (int k = 0; k < FDIM; ++k) s += psum[(size_t)g * FDIM + k] * Wlin[o * FDIM + k];
    out[tid] = s * inv + blin[o];
}

extern "C" void kernel_launch(void* const* d_in, const int* in_sizes, int n_in,
                              void* d_out, int out_size, void* d_ws, size_t ws_size,
                              hipStream_t stream) {
    const float*     x     = (const float*)d_in[0];
    const long long* ei    = (const long long*)d_in[1];
    const long long* batch = (const long long*)d_in[2];
    const float* Wrel[3]  = {(const float*)d_in[3], (const float*)d_in[6], (const float*)d_in[9]};
    const float* brel[3]  = {(const float*)d_in[4], (const float*)d_in[7], (const float*)d_in[10]};
    const float* Wroot[3] = {(const float*)d_in[5], (const float*)d_in[8], (const float*)d_in[11]};
    const float* Wlin = (const float*)d_in[12];
    const float* blin = (const float*)d_in[13];
    float* out = (float*)d_out;

    const int n_nodes  = in_sizes[0] / FDIM;
    const int n_edges  = in_sizes[1] / 2;
    const int n_graphs = out_size / 2;
    const int NF = n_nodes * FDIM;

    float* ws   = (float*)d_ws;
    float* agg  = ws;                 // NF floats
    float* h1   = ws + (size_t)NF;    // NF floats
    float* h2   = h1 + (size_t)NF;    // NF floats
    float* psum = h2 + (size_t)NF;    // n_graphs*FDIM
    float* cnt  = psum + (size_t)n_graphs * FDIM;

    const int B = 256;
    const int zgrid  = (NF + B - 1) / B;
    const int sgrid  = (n_edges * 16 + B - 1) / B;
    const int ggrid  = (n_nodes + 15) / 16;        // 1 block (4 waves) per 16 rows
    const int pgrid  = (n_nodes * 16 + B - 1) / B;

    // ---- layer 1: h1 = relu(conv(x))
    gnn_zero_kernel<<<zgrid, B, 0, stream>>>(agg, NF);
    gnn_scatter_kernel<<<sgrid, B, 0, stream>>>(x, ei, agg, n_edges);
    gnn_gemm_kernel<<<ggrid, 128, 0, stream>>>(agg, x, Wrel[0], brel[0], Wroot[0], h1, n_nodes, 1);
    // ---- layer 2: h2 = relu(conv(h1))
    gnn_zero_kernel<<<zgrid, B, 0, stream>>>(agg, NF);
    gnn_scatter_kernel<<<sgrid, B, 0, stream>>>(h1, ei, agg, n_edges);
    gnn_gemm_kernel<<<ggrid, 128, 0, stream>>>(agg, h1, Wrel[1], brel[1], Wroot[1], h2, n_nodes, 1);
    // ---- layer 3: h1 <- conv(h2) (no relu), reusing h1 as h3
    gnn_zero_kernel<<<zgrid, B, 0, stream>>>(agg, NF);
    gnn_scatter_kernel<<<sgrid, B, 0, stream>>>(h2, ei, agg, n_edges);
    gnn_gemm_kernel<<<ggrid, 128, 0, stream>>>(agg, h2, Wrel[2], brel[2], Wroot[2], h1, n_nodes, 0);
    // ---- mean pool + final linear
    const int pz = n_graphs * FDIM + n_graphs;
    gnn_zero_kernel<<<(pz + B - 1) / B, B, 0, stream>>>(psum, pz);
    gnn_pool_kernel<<<pgrid, B, 0, stream>>>(h1, batch, psum, cnt, n_nodes);
    gnn_final_kernel<<<(n_graphs * 2 + B - 1) / B, B, 0, stream>>>(psum, cnt, Wlin, blin, out, n_graphs);
}